// GraphSemanticLoss_32916629356852
// MI455X (gfx1250) — compile-verified
//
#include <hip/hip_runtime.h>

// CDNA5 / gfx1250 — wave32, WMMA f32 16x16x4 for the Gram matrix.

typedef float v2f __attribute__((ext_vector_type(2)));
typedef float v8f __attribute__((ext_vector_type(8)));

#define BSZ 512
#define DIM 512
#define MARGIN 0.5f
#define PD_EPS 1e-6f

// workspace layout (in floats)
#define WS_DT    0                    // [B*B]  dt matrix
#define WS_SQ    (BSZ * BSZ)          // [B]    row squared norms
#define WS_S     (WS_SQ + BSZ)        // [B]    row sums
#define WS_DSUM  (WS_S + BSZ)         // [128]  per-block sum(dist)
#define WS_DMAX  (WS_DSUM + 128)      // [128]  per-block max(dist)
#define WS_TRIP  (WS_DMAX + 128)      // [512]  per-anchor triplet sum
#define WS_CNT   (WS_TRIP + BSZ)      // [512]  per-anchor pair count (exact in f32)
#define WS_SIM   (WS_CNT + BSZ)       // [512]  per-anchor sum(sim row)

__device__ __forceinline__ float waveSum(float v) {
    #pragma unroll
    for (int o = 16; o > 0; o >>= 1) v += __shfl_down(v, o, 32);
    return v;
}
__device__ __forceinline__ float waveMax(float v) {
    #pragma unroll
    for (int o = 16; o > 0; o >>= 1) v = fmaxf(v, __shfl_down(v, o, 32));
    return v;
}
__device__ __forceinline__ int waveSumI(int v) {
    #pragma unroll
    for (int o = 16; o > 0; o >>= 1) v += __shfl_down(v, o, 32);
    return v;
}

// ---- kernel 1: per-row sum and squared-norm (wave per row) ----------------
__global__ __launch_bounds__(256) void row_stats_kernel(
    const float* __restrict__ E, float* __restrict__ ws) {
    const int lane = threadIdx.x & 31;
    const int wv   = threadIdx.x >> 5;
    const int row  = blockIdx.x * 8 + wv;          // 64 blocks * 8 waves = 512
    const float* p = E + row * DIM;
    float s = 0.f, q = 0.f;
    #pragma unroll 4
    for (int c = lane; c < DIM; c += 32) { float x = p[c]; s += x; q += x * x; }
    s = waveSum(s); q = waveSum(q);
    if (lane == 0) { ws[WS_S + row] = s; ws[WS_SQ + row] = q; }
}

// ---- kernel 2: Gram tile via V_WMMA_F32_16X16X4_F32 -> dist/dt ------------
// One wave per 16x16 output tile. 32x32 tiles, 8 waves/block, 128 blocks.
__global__ __launch_bounds__(256) void gram_dt_kernel(
    const float* __restrict__ E, float* __restrict__ ws) {
    __shared__ float red[16];

    const int lane = threadIdx.x & 31;
    const int wv   = threadIdx.x >> 5;
    const int tile = blockIdx.x * 8 + wv;          // 0..1023
    const int ti   = tile >> 5;                    // row tile
    const int tj   = tile & 31;                    // col tile
    const int i0   = ti * 16, j0 = tj * 16;
    const int r    = lane & 15;
    const int hi   = lane >> 4;                    // 0 or 1

    // f32 A 16x4: lane r = row, V0 holds K=2*hi, V1 holds K=2*hi+1.
    // f32 B 4x16 (= E^T tile): lane r = column -> same load pattern on E rows.
    const float* ar = E + (i0 + r) * DIM + 2 * hi;
    const float* br = E + (j0 + r) * DIM + 2 * hi;

    v8f c0 = {}; v8f c1 = {};                      // two accumulator chains
    for (int k0 = 0; k0 < DIM; k0 += 8) {
        v2f a0 = { ar[k0 + 0], ar[k0 + 1] };
        v2f b0 = { br[k0 + 0], br[k0 + 1] };
        v2f a1 = { ar[k0 + 4], ar[k0 + 5] };
        v2f b1 = { br[k0 + 4], br[k0 + 5] };
        c0 = __builtin_amdgcn_wmma_f32_16x16x4_f32(false, a0, false, b0,
                                                   (short)0, c0, false, false);
        c1 = __builtin_amdgcn_wmma_f32_16x16x4_f32(false, a1, false, b1,
                                                   (short)0, c1, false, false);
    }

    const float* sq = ws + WS_SQ;
    const float* sv = ws + WS_S;
    float*       dt = ws + WS_DT;

    const int   col = j0 + r;
    const float sqc = sq[col];
    const float sc  = sv[col];

    float dsum = 0.f, dmax = 0.f;
    #pragma unroll
    for (int v = 0; v < 8; v++) {
        // C/D layout: VGPR v, lanes 0-15 -> M=v, lanes 16-31 -> M=v+8
        const int   row = i0 + v + 8 * hi;
        const float g   = c0[v] + c1[v];
        float d2 = sq[row] + sqc - 2.0f * g;
        d2 = fmaxf(d2, 0.0f);
        const float dist = (d2 > 0.0f) ? sqrtf(d2) : 0.0f;
        // ||x + eps||^2 = d2 + 2*eps*(s_i - s_j) + D*eps^2
        const float dt2 = d2 + 2.0f * PD_EPS * (sv[row] - sc)
                             + (float)DIM * PD_EPS * PD_EPS;
        const float dtv = sqrtf(fmaxf(dt2, 0.0f));
        dt[row * BSZ + col] = dtv;
        dsum += dist;
        dmax = fmaxf(dmax, dist);
    }

    dsum = waveSum(dsum);
    dmax = waveMax(dmax);
    if (lane == 0) { red[wv] = dsum; red[8 + wv] = dmax; }
    __syncthreads();
    if (threadIdx.x == 0) {
        float ts = 0.f, tm = 0.f;
        #pragma unroll
        for (int w = 0; w < 8; w++) { ts += red[w]; tm = fmaxf(tm, red[8 + w]); }
        ws[WS_DSUM + blockIdx.x] = ts;
        ws[WS_DMAX + blockIdx.x] = tm;
    }
}

// ---- kernel 3: triplet cube slice per anchor (block per anchor) -----------
__global__ __launch_bounds__(256) void triplet_kernel(
    const float* __restrict__ sim, float* __restrict__ ws) {
    __shared__ float pdt[BSZ];     // dt if positive else -1e30 (relu -> 0)
    __shared__ float ndt[BSZ];     // dt if negative else +1e30 (relu -> 0)
    __shared__ float red[32];

    const int i = blockIdx.x;
    const float* srow = sim + i * BSZ;
    const float* drow = ws + WS_DT + i * BSZ;

    float simsum = 0.f;
    int npos = 0, nneg = 0;
    for (int c = threadIdx.x; c < BSZ; c += 256) {
        const float svv = srow[c];
        const float dvv = drow[c];
        simsum += svv;
        const bool p = svv > 0.7f;
        const bool n = svv < 0.3f;
        pdt[c] = p ? dvv : -1e30f;
        ndt[c] = n ? dvv :  1e30f;
        npos += p ? 1 : 0;
        nneg += n ? 1 : 0;
    }
    __syncthreads();

    float acc = 0.f;
    for (int j = 0; j < BSZ; j++) {
        const float a = pdt[j] + MARGIN;       // uniform across the block
        if (a < -1e29f) continue;              // non-positive anchor->j
        #pragma unroll 2
        for (int k = threadIdx.x; k < BSZ; k += 256)
            acc += fmaxf(a - ndt[k], 0.0f);
    }

    // deterministic block reduction of {acc, simsum, npos, nneg}
    const int lane = threadIdx.x & 31, wv = threadIdx.x >> 5;
    acc    = waveSum(acc);
    simsum = waveSum(simsum);
    npos   = waveSumI(npos);
    nneg   = waveSumI(nneg);
    if (lane == 0) {
        red[wv]      = acc;
        red[8 + wv]  = simsum;
        red[16 + wv] = (float)npos;
        red[24 + wv] = (float)nneg;
    }
    __syncthreads();
    if (threadIdx.x == 0) {
        float ta = 0.f, tsim = 0.f; int tp = 0, tn = 0;
        #pragma unroll
        for (int w = 0; w < 8; w++) {
            ta += red[w]; tsim += red[8 + w];
            tp += (int)red[16 + w]; tn += (int)red[24 + w];
        }
        ws[WS_TRIP + i] = ta;
        ws[WS_CNT  + i] = (float)(tp * tn);    // <= 262144, exact in f32
        ws[WS_SIM  + i] = tsim;
    }
}

// ---- kernel 4: final combine ----------------------------------------------
__global__ __launch_bounds__(256) void finalize_kernel(
    const float* __restrict__ ws, float* __restrict__ out) {
    __shared__ float red[40];
    const int tid = threadIdx.x, lane = tid & 31, wv = tid >> 5;

    float dsum = 0.f, dmax = 0.f, tsum = 0.f, csum = 0.f, ssum = 0.f;
    for (int idx = tid; idx < 128; idx += 256) {
        dsum += ws[WS_DSUM + idx];
        dmax = fmaxf(dmax, ws[WS_DMAX + idx]);
    }
    for (int idx = tid; idx < BSZ; idx += 256) {
        tsum += ws[WS_TRIP + idx];
        csum += ws[WS_CNT + idx];
        ssum += ws[WS_SIM + idx];
    }
    dsum = waveSum(dsum); dmax = waveMax(dmax);
    tsum = waveSum(tsum); csum = waveSum(csum); ssum = waveSum(ssum);
    if (lane == 0) {
        red[wv] = dsum; red[8 + wv] = dmax; red[16 + wv] = tsum;
        red[24 + wv] = csum; red[32 + wv] = ssum;
    }
    __syncthreads();
    if (tid == 0) {
        float ds = 0.f, dm = 0.f, ts = 0.f, cs = 0.f, ss = 0.f;
        #pragma unroll
        for (int w = 0; w < 8; w++) {
            ds += red[w]; dm = fmaxf(dm, red[8 + w]); ts += red[16 + w];
            cs += red[24 + w]; ss += red[32 + w];
        }
        const float N = (float)BSZ * (float)BSZ;
        // mean((1-S)*dn - S*(1-dn)) == mean(dist)/max(dist) - mean(S)
        const float sim_loss = ds / (dm * N) - ss / N;
        const float trip = (cs > 0.f) ? (ts / cs) : 0.f;
        out[0] = trip + sim_loss;
    }
}

extern "C" void kernel_launch(void* const* d_in, const int* in_sizes, int n_in,
                              void* d_out, int out_size, void* d_ws, size_t ws_size,
                              hipStream_t stream) {
    const float* E = (const float*)d_in[0];   // embeddings [512,512] f32
    const float* S = (const float*)d_in[1];   // similarity [512,512] f32
    float* ws  = (float*)d_ws;
    float* out = (float*)d_out;

    row_stats_kernel<<<64, 256, 0, stream>>>(E, ws);
    gram_dt_kernel  <<<128, 256, 0, stream>>>(E, ws);
    triplet_kernel  <<<BSZ, 256, 0, stream>>>(S, ws);
    finalize_kernel <<<1, 256, 0, stream>>>(ws, out);
}